// VCL_64321430225494
// MI455X (gfx1250) — compile-verified
//
#include <hip/hip_runtime.h>
#include <hip/hip_bf16.h>

#define BDIM 512
#define DDIM 256
#define BETA 1.0f

typedef __attribute__((ext_vector_type(2))) float v2f;
typedef __attribute__((ext_vector_type(8))) float v8f;

__device__ __forceinline__ float blockReduceSum(float v, float* smem) {
    #pragma unroll
    for (int off = 16; off > 0; off >>= 1) v += __shfl_down(v, off, 32);
    const int lane = threadIdx.x & 31, w = threadIdx.x >> 5;
    __syncthreads();
    if (lane == 0) smem[w] = v;
    __syncthreads();
    float r = 0.0f;
    if ((int)threadIdx.x < (int)(blockDim.x >> 5)) r = smem[threadIdx.x];
    if (w == 0) {
        #pragma unroll
        for (int off = 16; off > 0; off >>= 1) r += __shfl_down(r, off, 32);
    }
    return r; // valid in thread 0
}

__device__ __forceinline__ float blockReduceMax(float v, float* smem) {
    #pragma unroll
    for (int off = 16; off > 0; off >>= 1) v = fmaxf(v, __shfl_down(v, off, 32));
    const int lane = threadIdx.x & 31, w = threadIdx.x >> 5;
    __syncthreads();
    if (lane == 0) smem[w] = v;
    __syncthreads();
    float r = -3.402823466e+38f;
    if ((int)threadIdx.x < (int)(blockDim.x >> 5)) r = smem[threadIdx.x];
    if (w == 0) {
        #pragma unroll
        for (int off = 16; off > 0; off >>= 1) r = fmaxf(r, __shfl_down(r, off, 32));
    }
    return r; // valid in thread 0
}

// Kernel 1: per-row precompute. grid = B blocks of D threads.
//   IV[b,d]  = exp(-logvar)
//   W[b,d]   = -2 * mu * IV
//   crow[b]  = sum_d (mu^2 * IV + logvar)
//   klp[b]   = sum_d 0.5*(-1 - logvar + mu^2 + exp(logvar))
__global__ void __launch_bounds__(DDIM) vcl_prep(const float* __restrict__ mu,
                                                 const float* __restrict__ logvar,
                                                 float* __restrict__ IV,
                                                 float* __restrict__ W,
                                                 float* __restrict__ crow,
                                                 float* __restrict__ klp) {
    __shared__ float sred[32];
    const int b = blockIdx.x;
    const int d = threadIdx.x;
    const int idx = b * DDIM + d;
    const float lv = logvar[idx];
    const float m  = mu[idx];
    const float iv = __expf(-lv);
    IV[idx] = iv;
    W[idx]  = -2.0f * m * iv;
    const float cpart  = m * m * iv + lv;
    const float klpart = 0.5f * (-1.0f - lv + m * m + __expf(lv));
    const float cs = blockReduceSum(cpart, sred);
    __syncthreads();
    const float ks = blockReduceSum(klpart, sred);
    if (threadIdx.x == 0) { crow[b] = cs; klp[b] = ks; }
}

// Kernel 2: S[b,n] = -0.5*( IV[b,:]·z2sq[n,:] + W[b,:]·z2[n,:] + crow[b] )
// One wave per 16x16 tile, fp32 WMMA 16x16x4, two WMMAs per K-step.
__global__ void __launch_bounds__(256) vcl_gemm(const float* __restrict__ IV,
                                                const float* __restrict__ W,
                                                const float* __restrict__ z2,
                                                const float* __restrict__ crow,
                                                float* __restrict__ S) {
    const int wave  = (blockIdx.x * blockDim.x + threadIdx.x) >> 5;
    const int lane  = threadIdx.x & 31;
    const int tm    = wave >> 5;   // b-tile 0..31
    const int tn    = wave & 31;   // n-tile 0..31
    const int idx16 = lane & 15;
    const int half  = lane >> 4;

    // A fragment (16x4 fp32): lane%16 = M row; lanes 0-15 carry K={0,1},
    // lanes 16-31 carry K={2,3} in the two VGPRs.
    const float* ivp = IV + (tm * 16 + idx16) * DDIM;
    const float* wp  = W  + (tm * 16 + idx16) * DDIM;
    // B fragment (4x16 fp32), mirrored: lane%16 = N column, same K split.
    const float* zp  = z2 + (tn * 16 + idx16) * DDIM;

    v8f acc = {};
    #pragma unroll 4
    for (int k = 0; k < DDIM; k += 4) {
        const int kk = k + half * 2;
        const v2f a_iv = *(const v2f*)(ivp + kk);
        const v2f a_w  = *(const v2f*)(wp  + kk);
        const v2f b_z  = *(const v2f*)(zp  + kk);
        v2f b_zq; b_zq.x = b_z.x * b_z.x; b_zq.y = b_z.y * b_z.y;
        // acc += IV_tile * (z^2)_tile
        acc = __builtin_amdgcn_wmma_f32_16x16x4_f32(false, a_iv, false, b_zq,
                                                    (short)0, acc, false, false);
        // acc += (-2 mu iv)_tile * z_tile
        acc = __builtin_amdgcn_wmma_f32_16x16x4_f32(false, a_w, false, b_z,
                                                    (short)0, acc, false, false);
    }

    // C/D layout: VGPR r, lanes 0-15 -> (M=r, N=lane); lanes 16-31 -> (M=8+r, N=lane-16)
    const int n = tn * 16 + idx16;
    #pragma unroll
    for (int r = 0; r < 8; ++r) {
        const int m = tm * 16 + half * 8 + r;
        S[m * BDIM + n] = -0.5f * (acc[r] + crow[m]);
    }
}

// Kernel 3: logsumexp of each row (blocks 0..511) and each column (blocks 512..1023).
__global__ void __launch_bounds__(256) vcl_lse(const float* __restrict__ S,
                                               float* __restrict__ lse) {
    __shared__ float sred[32];
    __shared__ float smax;
    const int i = blockIdx.x;       // 0..1023
    const int t = threadIdx.x;      // 0..255, two elements each
    float v0, v1;
    if (i < BDIM) {
        const float* row = S + i * BDIM;
        v0 = row[t];
        v1 = row[t + 256];
    } else {
        const int c = i - BDIM;
        v0 = S[t * BDIM + c];
        v1 = S[(t + 256) * BDIM + c];
    }
    float mx = blockReduceMax(fmaxf(v0, v1), sred);
    if (t == 0) smax = mx;
    __syncthreads();
    const float m = smax;
    float s = __expf(v0 - m) + __expf(v1 - m);
    s = blockReduceSum(s, sred);
    if (t == 0) lse[i] = __logf(s) + m;
}

// Kernel 4: scalar fold. One block of 512 threads.
__global__ void __launch_bounds__(BDIM) vcl_final(const float* __restrict__ S,
                                                  const float* __restrict__ lse,
                                                  const float* __restrict__ klp,
                                                  float* __restrict__ out) {
    __shared__ float sred[32];
    const int t = threadIdx.x;      // 0..511
    const float diag = S[t * BDIM + t];
    const float v = (lse[t] - diag) + (lse[BDIM + t] - diag);
    const float k = klp[t];
    const float sv = blockReduceSum(v, sred);
    __syncthreads();
    const float sk = blockReduceSum(k, sred);
    if (t == 0) out[0] = sv * (1.0f / (float)BDIM) + BETA * (sk * (1.0f / (float)BDIM));
}

extern "C" void kernel_launch(void* const* d_in, const int* in_sizes, int n_in,
                              void* d_out, int out_size, void* d_ws, size_t ws_size,
                              hipStream_t stream) {
    const float* mu     = (const float*)d_in[0];
    const float* logvar = (const float*)d_in[1];
    const float* z2     = (const float*)d_in[2];
    float* out = (float*)d_out;

    // Workspace layout (floats):
    float* ws   = (float*)d_ws;
    float* IV   = ws;                         // 512*256
    float* W    = IV + BDIM * DDIM;           // 512*256
    float* S    = W  + BDIM * DDIM;           // 512*512
    float* crow = S  + BDIM * BDIM;           // 512
    float* klp  = crow + BDIM;                // 512
    float* lse  = klp  + BDIM;                // 1024

    vcl_prep<<<BDIM, DDIM, 0, stream>>>(mu, logvar, IV, W, crow, klp);
    // 1024 tiles (32x32 grid of 16x16), 8 waves per 256-thread block -> 128 blocks
    vcl_gemm<<<128, 256, 0, stream>>>(IV, W, z2, crow, S);
    vcl_lse<<<2 * BDIM, 256, 0, stream>>>(S, lse);
    vcl_final<<<1, BDIM, 0, stream>>>(S, lse, klp, out);
}